// StitchModel_44418551775296
// MI455X (gfx1250) — compile-verified
//
#include <hip/hip_runtime.h>

typedef __attribute__((ext_vector_type(2))) float v2f;
typedef __attribute__((ext_vector_type(8))) float v8f;

#define BIGF 3.402823e38f

// Branchless streaming top-2 insert (j visited in increasing order, so a
// strict < reproduces top_k's smaller-index tie-break).
__device__ __forceinline__ void ins_seq(float d, int j,
                                        float& b1, int& j1,
                                        float& b2, int& j2) {
    const bool lt1 = d < b1;
    const bool lt2 = d < b2;
    const float s2  = lt1 ? b1 : d;
    const int   sj2 = lt1 ? j1 : j;
    b2 = lt2 ? s2  : b2;
    j2 = lt2 ? sj2 : j2;
    b1 = lt1 ? d : b1;
    j1 = lt1 ? j : j1;
}

// Branchless insert with (d, j) lexicographic compare, for merging candidate
// sets whose indices interleave (cross-lane reduction).
__device__ __forceinline__ void ins_lex(float d, int j,
                                        float& b1, int& j1,
                                        float& b2, int& j2) {
    const bool lt1 = (d < b1) || (d == b1 && j < j1);
    const bool lt2 = (d < b2) || (d == b2 && j < j2);
    const float s2  = lt1 ? b1 : d;
    const int   sj2 = lt1 ? j1 : j;
    b2 = lt2 ? s2  : b2;
    j2 = lt2 ? sj2 : j2;
    b1 = lt1 ? d : b1;
    j1 = lt1 ? j : j1;
}

// One wave32 handles a 16-row tile of bbox1 and scans all of bbox2 in 16-col
// tiles. Squared distance for the whole 16x16 tile comes out of ONE
// v_wmma_f32_16x16x4_f32 via the embedding:
//   A[i] = (1, x1, y1, x1^2+y1^2),  B[j] = (x2^2+y2^2, -2*x2, -2*y2, 1)
//   => (A.B)[i][j] = (x1-x2)^2 + (y1-y2)^2
__global__ void __launch_bounds__(256)
topk_dist_kernel(const float* __restrict__ bbox1,
                 const float* __restrict__ bbox2,
                 int* __restrict__ nbr, int n1, int n2) {
    const int lane = threadIdx.x & 31;
    const int wave = threadIdx.x >> 5;
    const int tile = blockIdx.x * (blockDim.x >> 5) + wave;
    const int i0 = tile * 16;
    const int m = lane & 15;       // A-row for this lane / B-column for this lane
    const bool hi = lane >= 16;    // high half supplies K=2,3 of A and B

    // A operand: constant over the whole j loop.
    const float x1 = bbox1[(i0 + m) * 4 + 0];
    const float y1 = bbox1[(i0 + m) * 4 + 1];
    v2f a;
    a.x = hi ? y1 : 1.0f;                 // K=2 : K=0
    a.y = hi ? (x1 * x1 + y1 * y1) : x1;  // K=3 : K=1

    // Running top-2 (value,index) per C-slot r. Lane tracks rows
    // M = r (low half) / M = r+8 (high half), columns j == jt + m.
    float best1[8], best2[8];
    int   bj1[8],   bj2[8];
#pragma unroll
    for (int r = 0; r < 8; ++r) {
        best1[r] = BIGF; best2[r] = BIGF; bj1[r] = 0; bj2[r] = 0;
    }

#pragma unroll 2
    for (int jt = 0; jt < n2; jt += 16) {
        const int j = jt + m;
        const float x2 = bbox2[j * 4 + 0];
        const float y2 = bbox2[j * 4 + 1];
        v2f b;
        b.x = hi ? (-2.0f * y2) : (x2 * x2 + y2 * y2);  // K=2 : K=0
        b.y = hi ? 1.0f : (-2.0f * x2);                 // K=3 : K=1

        v8f c = {};
        c = __builtin_amdgcn_wmma_f32_16x16x4_f32(
                /*neg_a=*/false, a, /*neg_b=*/false, b,
                /*c_mod=*/(short)0, c, /*reuse_a=*/false, /*reuse_b=*/false);

#pragma unroll
        for (int r = 0; r < 8; ++r) {
            ins_seq(c[r], j, best1[r], bj1[r], best2[r], bj2[r]);
        }
    }

    // Merge top-2 across the 16 lanes of each half (xor masks < 16 never
    // cross the half boundary on wave32). Lexicographic compare reproduces
    // top_k's smaller-index tie-break.
#pragma unroll
    for (int mask = 1; mask <= 8; mask <<= 1) {
#pragma unroll
        for (int r = 0; r < 8; ++r) {
            const float o1  = __shfl_xor(best1[r], mask, 32);
            const int   oj1 = __shfl_xor(bj1[r],   mask, 32);
            const float o2  = __shfl_xor(best2[r], mask, 32);
            const int   oj2 = __shfl_xor(bj2[r],   mask, 32);
            ins_lex(o1, oj1, best1[r], bj1[r], best2[r], bj2[r]);
            ins_lex(o2, oj2, best1[r], bj1[r], best2[r], bj2[r]);
        }
    }

    if (m == 0) {  // lane 0 writes rows 0..7, lane 16 writes rows 8..15
#pragma unroll
        for (int r = 0; r < 8; ++r) {
            const int i = i0 + (hi ? 8 + r : r);
            nbr[i * 2 + 0] = bj1[r];
            nbr[i * 2 + 1] = bj2[r];
        }
    }
}

// Build edge_index (as floats; d_out is one flat float buffer) and edge_attr.
__global__ void __launch_bounds__(256)
edge_kernel(const float* __restrict__ bbox1, const float* __restrict__ bbox2,
            const float* __restrict__ zflow1, const float* __restrict__ zflow2,
            const int* __restrict__ nbr, float* __restrict__ out, int n1) {
    const int ne = n1 * 2;
    const int e = blockIdx.x * blockDim.x + threadIdx.x;
    if (e >= ne) return;
    const int i = e >> 1;
    const int j = nbr[e];

    out[e]      = (float)i;           // edge_index[0][e]
    out[ne + e] = (float)(j + n1);    // edge_index[1][e]

    float* attr = out + 2 * ne + e * 6;
    attr[0] = zflow1[i];
    attr[1] = zflow2[j];
    attr[2] = bbox1[i * 4 + 0] - bbox2[j * 4 + 0];
    attr[3] = bbox1[i * 4 + 1] - bbox2[j * 4 + 1];
    attr[4] = bbox1[i * 4 + 2] / bbox2[j * 4 + 2];
    attr[5] = bbox1[i * 4 + 3] / bbox2[j * 4 + 3];
}

extern "C" void kernel_launch(void* const* d_in, const int* in_sizes, int n_in,
                              void* d_out, int out_size, void* d_ws, size_t ws_size,
                              hipStream_t stream) {
    const float* bbox1  = (const float*)d_in[0];
    const float* bbox2  = (const float*)d_in[1];
    const float* zflow1 = (const float*)d_in[2];
    const float* zflow2 = (const float*)d_in[3];
    const int n1 = in_sizes[0] / 4;   // 16384
    const int n2 = in_sizes[1] / 4;   // 16384

    int*   nbr = (int*)d_ws;          // n1*2 ints = 128 KB scratch
    float* out = (float*)d_out;

    const int tiles = n1 / 16;                // 1024 row tiles
    const int wavesPerBlock = 8;              // 256 threads = 8 wave32s
    dim3 block(32 * wavesPerBlock);
    dim3 grid(tiles / wavesPerBlock);         // 128 blocks
    topk_dist_kernel<<<grid, block, 0, stream>>>(bbox1, bbox2, nbr, n1, n2);

    const int ne = n1 * 2;
    edge_kernel<<<(ne + 255) / 256, 256, 0, stream>>>(bbox1, bbox2, zflow1,
                                                      zflow2, nbr, out, n1);
}